// ConsistentEmbeddingLoss_26173530701952
// MI455X (gfx1250) — compile-verified
//
#include <hip/hip_runtime.h>
#include <hip/hip_bf16.h>
#include <math.h>

typedef __attribute__((ext_vector_type(2))) float v2f;
typedef __attribute__((ext_vector_type(8))) float v8f;

#define NPTS 4096
#define DIM  256
#define KACT 262   // D + 6
#define KPAD 264   // padded to multiple of 4 for 16x16x4 WMMA
#define THRV 0.05f

__device__ __constant__ int c_pi[3] = {0, 0, 1};
__device__ __constant__ int c_pj[3] = {1, 2, 2};

// ---------------- init: zero accumulators + output ----------------
__global__ void init_kernel(float* cnt, float* sdf, float* out, int out_size) {
    int t = threadIdx.x;
    if (t < 3) { cnt[t] = 0.0f; sdf[t] = 0.0f; }
    if (t < out_size) out[t] = 0.0f;
}

// ---------------- pad W1 (262x256) -> W1p (264x256, zero rows 262..263) ----
__global__ __launch_bounds__(256) void padw1_kernel(const float* __restrict__ W1,
                                                    float* __restrict__ W1p) {
    int row = blockIdx.x;        // 0..263
    int col = threadIdx.x;       // 0..255
    W1p[row * DIM + col] = (row < KACT) ? W1[row * DIM + col] : 0.0f;
}

// ---------------- directed nearest-neighbor search ----------------
// block = ((b*3 + p)*2 + dir)*16 + qtile ; 256 threads, 1 query each
__global__ __launch_bounds__(256) void nn_kernel(const float* __restrict__ pts,
                                                 int* __restrict__ nn,
                                                 float* __restrict__ mind) {
    __shared__ float4 keys[1024];
    int blk   = blockIdx.x;
    int qtile = blk & 15;
    int dir   = (blk >> 4) & 1;
    int p     = (blk >> 5) % 3;
    int b     = blk / 96;
    int va = dir == 0 ? c_pi[p] : c_pj[p];
    int vb = dir == 0 ? c_pj[p] : c_pi[p];
    int tid = threadIdx.x;
    int n   = qtile * 256 + tid;

    const float* q = pts + (((size_t)(b * 3 + va)) * NPTS + n) * 3;
    float qx = q[0], qy = q[1], qz = q[2];
    float qn = qx * qx + qy * qy + qz * qz;

    const float* kb = pts + ((size_t)(b * 3 + vb)) * NPTS * 3;
    float best = 3.4e38f;
    int   bi   = 0;
    for (int c0 = 0; c0 < NPTS; c0 += 1024) {
        __syncthreads();
        for (int i = tid; i < 1024; i += 256) {
            const float* kp = kb + (size_t)(c0 + i) * 3;
            float kx = kp[0], ky = kp[1], kz = kp[2];
            keys[i] = make_float4(kx, ky, kz, kx * kx + ky * ky + kz * kz);
        }
        __syncthreads();
        for (int i = 0; i < 1024; ++i) {
            float4 k = keys[i];
            float d2 = qn + k.w - 2.0f * (qx * k.x + qy * k.y + qz * k.z);
            d2 = fmaxf(d2, 0.0f);                 // match reference clamp-before-argmin
            if (d2 < best) { best = d2; bi = c0 + i; }  // strict < keeps first index
        }
    }
    nn[((size_t)(b * 3 + p) * 2 + dir) * NPTS + n] = bi;
    if (dir == 0) mind[(size_t)(b * 3 + p) * NPTS + n] = sqrtf(best);
}

// ---------------- validity flags + per-pair counts ----------------
__global__ __launch_bounds__(256) void vc_kernel(const int* __restrict__ nn,
                                                 const float* __restrict__ mind,
                                                 const unsigned char* __restrict__ vm,
                                                 float* __restrict__ vc,
                                                 float* __restrict__ cnt) {
    __shared__ float red[256];
    int blk  = blockIdx.x;          // (b*3+p)*16 + tile
    int tile = blk & 15;
    int p    = (blk >> 4) % 3;
    int b    = blk / 48;
    int vi = c_pi[p], vj = c_pj[p];
    int tid = threadIdx.x;
    int n   = tile * 256 + tid;
    size_t base = (size_t)(b * 3 + p);

    int j     = nn[(base * 2 + 0) * NPTS + n];
    int back  = nn[(base * 2 + 1) * NPTS + j];
    bool mutual = (back == n);
    bool thr    = mind[base * NPTS + n] < THRV;
    bool mi     = vm[(size_t)(b * 3 + vi) * NPTS + n] != 0;
    bool mj     = vm[(size_t)(b * 3 + vj) * NPTS + j] != 0;
    float v = (mutual && thr && mi && mj) ? 1.0f : 0.0f;
    vc[base * NPTS + n] = v;

    red[tid] = v;
    __syncthreads();
    for (int s = 128; s; s >>= 1) {
        if (tid < s) red[tid] += red[tid + s];
        __syncthreads();
    }
    if (tid == 0) atomicAdd(&cnt[p], red[0]);
}

// ---------------- fused MLP (WMMA f32) + masked squared-diff ----------------
// block = b*768 + p*256 + rowtile ; 256 threads = 8 waves; 16 rows per block
__global__ __launch_bounds__(256) void transform_kernel(
    const float* __restrict__ emb, const float* __restrict__ pts,
    const float* __restrict__ W1p, const float* __restrict__ b1,
    const float* __restrict__ lng, const float* __restrict__ lnb,
    const float* __restrict__ W2,  const float* __restrict__ b2,
    const int* __restrict__ nn,    const float* __restrict__ vc,
    float* __restrict__ sumdiff) {

    __shared__ float xs[16][KPAD + 1];   // x tile (16 x 264), +1 anti-conflict pad
    __shared__ float hs[16][DIM + 1];    // hidden tile
    __shared__ float ej[16][DIM + 1];    // gathered target embeddings
    __shared__ float vcs[16];
    __shared__ int   js[16];
    __shared__ float red[256];

    int blk  = blockIdx.x;
    int tile = blk & 255;
    int p    = (blk >> 8) % 3;
    int b    = blk / 768;
    int vi = c_pi[p], vj = c_pj[p];
    int rowbase = tile * 16;
    int tid = threadIdx.x;

    // per-row metadata: NN index, vc flag, rel + rdir into x cols 256..263
    if (tid < 16) {
        int n = rowbase + tid;
        int j = nn[((size_t)(b * 3 + p) * 2 + 0) * NPTS + n];
        js[tid]  = j;
        vcs[tid] = vc[(size_t)(b * 3 + p) * NPTS + n];
        const float* pin = pts + (((size_t)(b * 3 + vi)) * NPTS + n) * 3;
        const float* pjn = pts + (((size_t)(b * 3 + vj)) * NPTS + j) * 3;
        float rx = pjn[0] - pin[0], ry = pjn[1] - pin[1], rz = pjn[2] - pin[2];
        float nrm = sqrtf(rx * rx + ry * ry + rz * rz);
        float inv = 1.0f / fmaxf(nrm, 1e-6f);
        xs[tid][256] = rx;       xs[tid][257] = ry;       xs[tid][258] = rz;
        xs[tid][259] = rx * inv; xs[tid][260] = ry * inv; xs[tid][261] = rz * inv;
        xs[tid][262] = 0.0f;     xs[tid][263] = 0.0f;
    }
    __syncthreads();

    // bulk load emb_i rows -> xs[:,0:256], gathered emb_j -> ej
    const float* ebase_i = emb + (((size_t)(b * 3 + vi)) * NPTS + rowbase) * DIM;
    const float* ebase_j = emb + ((size_t)(b * 3 + vj)) * NPTS * DIM;
    for (int it = 0; it < 4; ++it) {
        int v = tid + it * 256;          // 0..1023 float4 slots (16 rows x 64)
        int r = v >> 6;
        int c = (v & 63) << 2;
        float4 e = *(const float4*)(ebase_i + (size_t)r * DIM + c);
        xs[r][c] = e.x; xs[r][c + 1] = e.y; xs[r][c + 2] = e.z; xs[r][c + 3] = e.w;
        float4 f = *(const float4*)(ebase_j + (size_t)js[r] * DIM + c);
        ej[r][c] = f.x; ej[r][c + 1] = f.y; ej[r][c + 2] = f.z; ej[r][c + 3] = f.w;
    }
    __syncthreads();

    int wave  = tid >> 5;
    int lane  = tid & 31;
    int mrow  = lane & 15;                 // A: M index
    int khalf = (lane >> 4) << 1;          // A/B: K half (0 or 2)
    int mbase = (lane >> 4) << 3;          // C/D: M base (0 or 8)

    // ---- GEMM1: h = x @ W1p + b1, K = 264 via V_WMMA_F32_16X16X4_F32 ----
    for (int t = 0; t < 2; ++t) {
        int col0 = (wave + t * 8) * 16;
        int ncol = col0 + (lane & 15);
        v8f acc = {};
        for (int kk = 0; kk < KPAD; kk += 4) {
            int k0 = kk + khalf;
            v2f a, bv;
            a.x  = xs[mrow][k0];
            a.y  = xs[mrow][k0 + 1];
            bv.x = W1p[(size_t)k0 * DIM + ncol];
            bv.y = W1p[(size_t)(k0 + 1) * DIM + ncol];
            acc = __builtin_amdgcn_wmma_f32_16x16x4_f32(
                false, a, false, bv, (short)0, acc, false, false);
        }
        float bb = b1[ncol];
#pragma unroll
        for (int r = 0; r < 8; ++r) hs[mbase + r][ncol] = acc[r] + bb;
    }
    __syncthreads();

    // ---- LayerNorm + exact GELU per row (2 rows per wave) ----
    for (int rr = 0; rr < 2; ++rr) {
        int r = wave * 2 + rr;
        float s = 0.0f;
        for (int c = lane; c < DIM; c += 32) s += hs[r][c];
        for (int off = 16; off; off >>= 1) s += __shfl_xor(s, off, 32);
        float mu = s * (1.0f / 256.0f);
        float vv = 0.0f;
        for (int c = lane; c < DIM; c += 32) { float d = hs[r][c] - mu; vv += d * d; }
        for (int off = 16; off; off >>= 1) vv += __shfl_xor(vv, off, 32);
        float rstd = rsqrtf(vv * (1.0f / 256.0f) + 1e-5f);
        for (int c = lane; c < DIM; c += 32) {
            float hn = (hs[r][c] - mu) * rstd * lng[c] + lnb[c];
            hs[r][c] = 0.5f * hn * (1.0f + erff(hn * 0.70710678118654752f));
        }
    }
    __syncthreads();

    // ---- GEMM2: et = g @ W2 + b2, then masked squared diff ----
    float tsum = 0.0f;
    for (int t = 0; t < 2; ++t) {
        int col0 = (wave + t * 8) * 16;
        int ncol = col0 + (lane & 15);
        v8f acc = {};
        for (int kk = 0; kk < DIM; kk += 4) {
            int k0 = kk + khalf;
            v2f a, bv;
            a.x  = hs[mrow][k0];
            a.y  = hs[mrow][k0 + 1];
            bv.x = W2[(size_t)k0 * DIM + ncol];
            bv.y = W2[(size_t)(k0 + 1) * DIM + ncol];
            acc = __builtin_amdgcn_wmma_f32_16x16x4_f32(
                false, a, false, bv, (short)0, acc, false, false);
        }
        float bb = b2[ncol];
#pragma unroll
        for (int r = 0; r < 8; ++r) {
            int m = mbase + r;
            float d = (acc[r] + bb) - ej[m][ncol];
            tsum += d * d * vcs[m];
        }
    }
    red[tid] = tsum;
    __syncthreads();
    for (int s = 128; s; s >>= 1) {
        if (tid < s) red[tid] += red[tid + s];
        __syncthreads();
    }
    if (tid == 0) atomicAdd(&sumdiff[p], red[0]);
}

// ---------------- finalize scalar loss ----------------
__global__ void fin_kernel(const float* cnt, const float* sdf, float* out) {
    float total = 0.0f, np = 0.0f;
    for (int p = 0; p < 3; ++p) {
        float c = cnt[p];
        if (c > 0.0f) {
            total += sdf[p] / (c * 256.0f + 1e-6f);
            np += 1.0f;
        }
    }
    out[0] = (np > 0.0f) ? total / np : 0.0f;
}

extern "C" void kernel_launch(void* const* d_in, const int* in_sizes, int n_in,
                              void* d_out, int out_size, void* d_ws, size_t ws_size,
                              hipStream_t stream) {
    const float*         emb = (const float*)d_in[0];          // (2,3,4096,256)
    const float*         pts = (const float*)d_in[1];          // (2,3,4096,3)
    const unsigned char* vm  = (const unsigned char*)d_in[2];  // (2,3,4096) bool
    const float*         W1  = (const float*)d_in[3];          // (262,256)
    const float*         b1  = (const float*)d_in[4];
    const float*         lng = (const float*)d_in[5];
    const float*         lnb = (const float*)d_in[6];
    const float*         W2  = (const float*)d_in[7];          // (256,256)
    const float*         b2  = (const float*)d_in[8];
    float* out = (float*)d_out;

    // workspace layout (4-byte units)
    float* cnt  = (float*)d_ws;                       // 3
    float* sdf  = cnt + 4;                            // 3 (padded)
    float* W1p  = sdf + 4;                            // 264*256
    int*   nn   = (int*)(W1p + KPAD * DIM);           // 2*3*2*4096
    float* mind = (float*)(nn + 2 * 3 * 2 * NPTS);    // 2*3*4096
    float* vc   = mind + 2 * 3 * NPTS;                // 2*3*4096

    init_kernel<<<1, 256, 0, stream>>>(cnt, sdf, out, out_size);
    padw1_kernel<<<KPAD, 256, 0, stream>>>(W1, W1p);
    nn_kernel<<<2 * 3 * 2 * 16, 256, 0, stream>>>(pts, nn, mind);
    vc_kernel<<<2 * 3 * 16, 256, 0, stream>>>(nn, mind, vm, vc, cnt);
    transform_kernel<<<2 * 3 * 256, 256, 0, stream>>>(emb, pts, W1p, b1, lng, lnb,
                                                      W2, b2, nn, vc, sdf);
    fin_kernel<<<1, 1, 0, stream>>>(cnt, sdf, out);
}